// CNF_10874857194093
// MI455X (gfx1250) — compile-verified
//
#include <hip/hip_runtime.h>
#include <math.h>

typedef __attribute__((ext_vector_type(2))) float v2f;
typedef __attribute__((ext_vector_type(8))) float v8f;

#define B_N 2048
#define D_N 64
#define H_N 512

// ---------------------------------------------------------------------------
// f32 WMMA 16x16x4:  D = A(16x4) * B(4x16) + C(16x16), full f32 precision.
// ---------------------------------------------------------------------------
__device__ __forceinline__ v8f wmma_f32_k4(v2f a, v2f b, v8f c) {
  // 8 args: (neg_a, A, neg_b, B, c_mod, C, reuse_a, reuse_b)
  return __builtin_amdgcn_wmma_f32_16x16x4_f32(
      false, a, false, b, (short)0, c, false, false);
}

// ---------------------------------------------------------------------------
// One wave computes a 16x64 strip: 4 adjacent 16x16 tiles with 4 independent
// accumulators (keeps the matrix pipe fed, no D->C back-to-back chain) and a
// shared A fragment (2.5 dwords loaded per WMMA instead of 4).
// Per the CDNA5 ISA VGPR layouts (CDNA5 ISA):
//   A lane element: (M = lane%16, K = kb + 2*(lane/16) + r)  -> contiguous pair
//   B lane element: (K = kb + 2*(lane/16) + r, N = lane%16)
//   C lane element: (M = r + 8*(lane/16),      N = lane%16)
// ---------------------------------------------------------------------------
template <int K>
__device__ __forceinline__ void gemm_strip_f32x4(const float* __restrict__ A, int lda,
                                                 const float* __restrict__ Bm, int ldb,
                                                 int m0, int n0, v8f c[4]) {
  const int lane = threadIdx.x & 31;
  const int hi = lane >> 4;
  const int lo = lane & 15;
  const float* __restrict__ arow = A + (size_t)(m0 + lo) * lda + 2 * hi;
  const float* __restrict__ bcol = Bm + (size_t)(2 * hi) * ldb + n0 + lo;
#pragma unroll 4
  for (int kb = 0; kb < K; kb += 4) {
    v2f a;
    a.x = arow[kb];
    a.y = arow[kb + 1];
    v2f b0, b1, b2, b3;
    const float* bp = bcol + (size_t)kb * ldb;
    b0.x = bp[0];       b0.y = bp[ldb];
    b1.x = bp[16];      b1.y = bp[ldb + 16];
    b2.x = bp[32];      b2.y = bp[ldb + 32];
    b3.x = bp[48];      b3.y = bp[ldb + 48];
    c[0] = wmma_f32_k4(a, b0, c[0]);
    c[1] = wmma_f32_k4(a, b1, c[1]);
    c[2] = wmma_f32_k4(a, b2, c[2]);
    c[3] = wmma_f32_k4(a, b3, c[3]);
  }
}

// ---------------------------------------------------------------------------
// Layer 1: A1 = tanh( x[:, :64] @ W1[:64, :] + t*W1[64, :] + b1 )
// ---------------------------------------------------------------------------
__global__ __launch_bounds__(32) void k_layer1(const float* __restrict__ x,
                                               const float* __restrict__ W1,
                                               const float* __restrict__ b1,
                                               const float* __restrict__ tptr,
                                               float* __restrict__ A1) {
  const int n0 = blockIdx.x * 64;
  const int m0 = blockIdx.y * 16;
  const int lane = threadIdx.x & 31;
  const int hi = lane >> 4, lo = lane & 15;
  v8f c[4] = {};
  gemm_strip_f32x4<D_N>(x, D_N + 1, W1, H_N, m0, n0, c);
  const float t = tptr[0];
#pragma unroll
  for (int j = 0; j < 4; ++j) {
    const int n = n0 + 16 * j + lo;
    const float add = t * W1[(size_t)D_N * H_N + n] + b1[n];
#pragma unroll
    for (int r = 0; r < 8; ++r) {
      A1[(size_t)(m0 + r + 8 * hi) * H_N + n] = tanhf(c[j][r] + add);
    }
  }
}

// ---------------------------------------------------------------------------
// Q[p,q] = W2[p,q] * C[q,p],  C = W3 @ W1[:64, :]   (sample-independent)
// Tile rows (M) = q over W3, cols (N) = p over W1 columns; transposed store.
// ---------------------------------------------------------------------------
__global__ __launch_bounds__(32) void k_qmat(const float* __restrict__ W1,
                                             const float* __restrict__ W2,
                                             const float* __restrict__ W3,
                                             float* __restrict__ Q) {
  const int n0 = blockIdx.x * 64;  // p
  const int m0 = blockIdx.y * 16;  // q
  const int lane = threadIdx.x & 31;
  const int hi = lane >> 4, lo = lane & 15;
  v8f c[4] = {};
  gemm_strip_f32x4<D_N>(W3, D_N, W1, H_N, m0, n0, c);
#pragma unroll
  for (int j = 0; j < 4; ++j) {
    const int p = n0 + 16 * j + lo;
#pragma unroll
    for (int r = 0; r < 8; ++r) {
      const int q = m0 + r + 8 * hi;
      Q[(size_t)p * H_N + q] = W2[(size_t)p * H_N + q] * c[j][r];
    }
  }
}

// ---------------------------------------------------------------------------
// Layer 2: A2 = tanh( A1 @ W2 + b2 )
// ---------------------------------------------------------------------------
__global__ __launch_bounds__(32) void k_layer2(const float* __restrict__ A1,
                                               const float* __restrict__ W2,
                                               const float* __restrict__ b2,
                                               float* __restrict__ A2) {
  const int n0 = blockIdx.x * 64;
  const int m0 = blockIdx.y * 16;
  const int lane = threadIdx.x & 31;
  const int hi = lane >> 4, lo = lane & 15;
  v8f c[4] = {};
  gemm_strip_f32x4<H_N>(A1, H_N, W2, H_N, m0, n0, c);
#pragma unroll
  for (int j = 0; j < 4; ++j) {
    const int n = n0 + 16 * j + lo;
    const float bb = b2[n];
#pragma unroll
    for (int r = 0; r < 8; ++r) {
      A2[(size_t)(m0 + r + 8 * hi) * H_N + n] = tanhf(c[j][r] + bb);
    }
  }
}

// ---------------------------------------------------------------------------
// Layer 3: out[:, 0:64] = (A2 @ W3 + b3) / 2    (one wave per 16-sample row
// strip: N=64 is exactly one 4-tile strip)
// ---------------------------------------------------------------------------
__global__ __launch_bounds__(32) void k_layer3(const float* __restrict__ A2,
                                               const float* __restrict__ W3,
                                               const float* __restrict__ b3,
                                               float* __restrict__ out) {
  const int m0 = blockIdx.x * 16;
  const int lane = threadIdx.x & 31;
  const int hi = lane >> 4, lo = lane & 15;
  v8f c[4] = {};
  gemm_strip_f32x4<H_N>(A2, H_N, W3, D_N, m0, 0, c);
#pragma unroll
  for (int j = 0; j < 4; ++j) {
    const int n = 16 * j + lo;
    const float bb = b3[n];
#pragma unroll
    for (int r = 0; r < 8; ++r) {
      out[(size_t)(m0 + r + 8 * hi) * (D_N + 1) + n] = 0.5f * (c[j][r] + bb);
    }
  }
}

// ---------------------------------------------------------------------------
// Divergence partials:  S = (1-A1^2) @ Q  (WMMA strip), fused with the
// (1-A2^2) column weighting and a 16-lane shuffle row-reduction.
// The u = 1-a1^2 A-fragment is computed once and shared by 4 accumulators.
// grid.x = 8 column groups (one 64-wide strip each), grid.y = sample tiles.
// ---------------------------------------------------------------------------
__global__ __launch_bounds__(32) void k_div_part(const float* __restrict__ A1,
                                                 const float* __restrict__ A2,
                                                 const float* __restrict__ Q,
                                                 float* __restrict__ part) {
  const int g = blockIdx.x;        // column group 0..7
  const int m0 = blockIdx.y * 16;  // sample tile
  const int n0 = g * 64;
  const int lane = threadIdx.x & 31;
  const int hi = lane >> 4, lo = lane & 15;

  const float* __restrict__ urow = A1 + (size_t)(m0 + lo) * H_N + 2 * hi;
  const float* __restrict__ qcol = Q + (size_t)(2 * hi) * H_N + n0 + lo;

  v8f c[4] = {};
#pragma unroll 4
  for (int kb = 0; kb < H_N; kb += 4) {
    const float a0 = urow[kb];
    const float a1 = urow[kb + 1];
    v2f a;
    a.x = 1.f - a0 * a0;  // u = 1 - tanh^2, computed once per K-step
    a.y = 1.f - a1 * a1;
    const float* bp = qcol + (size_t)kb * H_N;
    v2f b0, b1, b2, b3;
    b0.x = bp[0];       b0.y = bp[H_N];
    b1.x = bp[16];      b1.y = bp[H_N + 16];
    b2.x = bp[32];      b2.y = bp[H_N + 32];
    b3.x = bp[48];      b3.y = bp[H_N + 48];
    c[0] = wmma_f32_k4(a, b0, c[0]);
    c[1] = wmma_f32_k4(a, b1, c[1]);
    c[2] = wmma_f32_k4(a, b2, c[2]);
    c[3] = wmma_f32_k4(a, b3, c[3]);
  }

  float p[8];
#pragma unroll
  for (int r = 0; r < 8; ++r) p[r] = 0.f;
#pragma unroll
  for (int j = 0; j < 4; ++j) {
    const int n = n0 + 16 * j + lo;
#pragma unroll
    for (int r = 0; r < 8; ++r) {
      const float av = A2[(size_t)(m0 + r + 8 * hi) * H_N + n];
      p[r] += c[j][r] * (1.f - av * av);  // weight by v = 1 - tanh^2
    }
  }

  // reduce over the 16 lanes sharing the same C-row (lanes 0-15 -> M=r,
  // lanes 16-31 -> M=r+8); wave32, width-16 xor shuffle.
#pragma unroll
  for (int r = 0; r < 8; ++r) {
    float v = p[r];
    for (int off = 8; off; off >>= 1) v += __shfl_xor(v, off, 16);
    if (lo == 0) part[(size_t)(m0 + r + 8 * hi) * 8 + g] = v;
  }
}

// ---------------------------------------------------------------------------
// Final: out[:, 64] = 0.5 * sum of 8 partials (deterministic, no atomics)
// ---------------------------------------------------------------------------
__global__ void k_div_final(const float* __restrict__ part,
                            float* __restrict__ out) {
  const int b = blockIdx.x * blockDim.x + threadIdx.x;
  if (b < B_N) {
    float s = 0.f;
#pragma unroll
    for (int g = 0; g < 8; ++g) s += part[(size_t)b * 8 + g];
    out[(size_t)b * (D_N + 1) + D_N] = 0.5f * s;
  }
}

extern "C" void kernel_launch(void* const* d_in, const int* in_sizes, int n_in,
                              void* d_out, int out_size, void* d_ws, size_t ws_size,
                              hipStream_t stream) {
  (void)in_sizes; (void)n_in; (void)out_size; (void)ws_size;
  const float* t  = (const float*)d_in[0];
  const float* x  = (const float*)d_in[1];
  const float* W1 = (const float*)d_in[2];
  const float* b1 = (const float*)d_in[3];
  const float* W2 = (const float*)d_in[4];
  const float* b2 = (const float*)d_in[5];
  const float* W3 = (const float*)d_in[6];
  const float* b3 = (const float*)d_in[7];
  float* out = (float*)d_out;

  // workspace layout (floats): A1 | A2 | Q | part  (~9.6 MB total)
  float* A1   = (float*)d_ws;
  float* A2   = A1 + (size_t)B_N * H_N;
  float* Q    = A2 + (size_t)B_N * H_N;
  float* part = Q  + (size_t)H_N * H_N;

  const dim3 wave(32);
  k_layer1 <<<dim3(H_N / 64, B_N / 16), wave, 0, stream>>>(x, W1, b1, t, A1);
  k_qmat   <<<dim3(H_N / 64, H_N / 16), wave, 0, stream>>>(W1, W2, W3, Q);
  k_layer2 <<<dim3(H_N / 64, B_N / 16), wave, 0, stream>>>(A1, W2, b2, A2);
  k_layer3 <<<dim3(B_N / 16), wave, 0, stream>>>(A2, W3, b3, out);
  k_div_part<<<dim3(8, B_N / 16), wave, 0, stream>>>(A1, A2, Q, part);
  k_div_final<<<dim3(B_N / 256), dim3(256), 0, stream>>>(part, out);
}